// SparseLinear_68771016343946
// MI455X (gfx1250) — compile-verified
//
#include <hip/hip_runtime.h>

// ---------------------------------------------------------------------------
// SparseLinear (y = x @ W^T + b), W = coalesced COO sorted by (row, col).
//   x: [8192, 20000] f32   w: [600000] f32   bias: [3000] f32
//   rows/cols: [600000] i32 (rows ascending)   y: [8192, 3000] f32
//
// Memory-bound: min traffic ≈ 655 MB (x) + 98 MB (y) + 7 MB (COO) ≈ 760 MB
// → ~33 us @ 23.3 TB/s. Compute (~157 GFLOP through the matrix pipes at 16x
// row-slot inflation) is negligible, so the design goal is: stream x from HBM
// exactly once, serve the ~30x reuse of each x element from the 192 MB L2,
// and express the accumulation via V_WMMA_F32_16X16X4_F32 with a minimal
// per-WMMA instruction count.
//
// Mapping per wave (wave32): C[16 batch x 16 rows] f32 accumulator (v8f).
// Each WMMA consumes 4 consecutive nnz of the 16-row tile:
//   A (16x4)  = gathered x[b, col_k]   (lanes 0-15: K{0,1}, 16-31: K{2,3})
//   B (4x16)  = weight scattered to its row slot (B[k][n]=w_k iff row_k==n)
// COO entries are packed in LDS as uint2 {row<<16|col, w}; a lane's two
// K-slots are adjacent, so one ds_load_b128 fetches both. The staged block is
// padded to a multiple of 8 with {row=31, w=0} sentinels → the inner loop is
// branch-free and EXEC stays all-1s around every WMMA.
// ---------------------------------------------------------------------------

typedef float v2f __attribute__((ext_vector_type(2)));
typedef float v8f __attribute__((ext_vector_type(8)));

#define IN_F   20000
#define OUT_F  3000
#define BATCHN 8192

#define ROW_TILE 16
#define WAVES    8                       // 256 threads = 8 wave32
#define TPB      (WAVES * 32)
#define BATCH_PER_BLOCK (WAVES * 16)     // each wave owns a 16-batch sub-tile
#define STAGE    512                     // nnz staged in LDS per sweep (4 KB)

__device__ __forceinline__ v8f chunk_mma(const float* __restrict__ xrow,
                                         unsigned n_idx, int k_hi,
                                         const uint2* s_pw,
                                         int c, v8f acc) {
  // Two adjacent packed records for this lane's K-slot pair; c and k_hi are
  // even-aligned so this is one aligned ds_load_b128 (LDS broadcast).
  const uint4 d = *(const uint4*)(&s_pw[c + 2 * k_hi]);

  v2f a, b;
  a.x = xrow[d.x & 0xFFFFu];                                  // gather x[b,col]
  a.y = xrow[d.z & 0xFFFFu];
  b.x = ((d.x >> 16) == n_idx) ? __uint_as_float(d.y) : 0.0f; // scatter weight
  b.y = ((d.z >> 16) == n_idx) ? __uint_as_float(d.w) : 0.0f;

  return __builtin_amdgcn_wmma_f32_16x16x4_f32(
      /*neg_a=*/false, a, /*neg_b=*/false, b,
      /*c_mod=*/(short)0, acc, /*reuse_a=*/false, /*reuse_b=*/false);
}

__global__ __launch_bounds__(TPB)
void spmm_wmma_kernel(const float* __restrict__ x,
                      const float* __restrict__ w,
                      const float* __restrict__ bias,
                      const int*   __restrict__ rows,
                      const int*   __restrict__ cols,
                      int nnz,
                      float* __restrict__ y) {
  __shared__ __align__(16) uint2 s_pw[STAGE];   // {row<<16 | col, w bits}
  __shared__ int s_range[2];

  const int tid  = threadIdx.x;
  const int lane = tid & 31;
  const int wave = tid >> 5;

  const int R  = blockIdx.x * ROW_TILE;                       // out-row tile
  const int B0 = blockIdx.y * BATCH_PER_BLOCK + wave * 16;    // batch sub-tile

  // COO is row-sorted: locate [row_ptr[R], row_ptr[R+16]) by binary search.
  if (tid < 2) {
    const int target = R + tid * ROW_TILE;
    int lo = 0, hi = nnz;
    while (lo < hi) {
      int mid = (lo + hi) >> 1;
      if (rows[mid] < target) lo = mid + 1; else hi = mid;
    }
    s_range[tid] = lo;
  }
  __syncthreads();
  const int k_begin = s_range[0];
  const int k_end   = s_range[1];

  v8f acc0 = {0.f, 0.f, 0.f, 0.f, 0.f, 0.f, 0.f, 0.f};
  v8f acc1 = {0.f, 0.f, 0.f, 0.f, 0.f, 0.f, 0.f, 0.f};

  const unsigned n_idx = (unsigned)(lane & 15);  // N slot == M (batch) slot
  const int      k_hi  = lane >> 4;              // 0 -> K{0,1}, 1 -> K{2,3}
  const float* __restrict__ xrow = x + (size_t)(B0 + (int)n_idx) * IN_F;

  for (int base = k_begin; base < k_end; base += STAGE) {
    const int cnt     = min(STAGE, k_end - base);
    const int cnt_pad = (cnt + 7) & ~7;          // pad to whole 2x(K=4) steps

    // Cooperative stage + sentinel padding (row=31 never matches, w=0).
    for (int i = tid; i < cnt_pad; i += TPB) {
      uint2 e;
      if (i < cnt) {
        e.x = ((unsigned)(rows[base + i] - R) << 16) | (unsigned)cols[base + i];
        e.y = __float_as_uint(w[base + i]);
      } else {
        e.x = 31u << 16;
        e.y = 0u;
      }
      s_pw[i] = e;
    }
    // Prefetch the next staged chunk while this one is consumed.
    const int nxt = base + STAGE + tid;
    if (nxt < k_end) {
      __builtin_prefetch(cols + nxt, 0, 1);
      __builtin_prefetch(w + nxt, 0, 1);
    }
    __syncthreads();

    // Branch-free inner loop; two accumulators hide WMMA->WMMA RAW latency.
    for (int c = 0; c < cnt_pad; c += 8) {
      acc0 = chunk_mma(xrow, n_idx, k_hi, s_pw, c,     acc0);
      acc1 = chunk_mma(xrow, n_idx, k_hi, s_pw, c + 4, acc1);
    }
    __syncthreads();
  }

  const v8f acc = acc0 + acc1;

  // C layout: vgpr v -> (M = v + 8*(lane>=16), N = lane&15).
  const int r = R + (int)n_idx;
  if (r < OUT_F) {                       // last tile holds rows 2992..2999 only
    const float bv = bias[r];
    const int m_base = (lane < 16) ? 0 : 8;
    #pragma unroll
    for (int v = 0; v < 8; ++v) {
      const int b = B0 + m_base + v;
      y[(size_t)b * OUT_F + r] = acc[v] + bv;    // 64B-contiguous per half-wave
    }
  }
}

extern "C" void kernel_launch(void* const* d_in, const int* in_sizes, int n_in,
                              void* d_out, int out_size, void* d_ws, size_t ws_size,
                              hipStream_t stream) {
  const float* x    = (const float*)d_in[0];
  const float* w    = (const float*)d_in[1];
  const float* bias = (const float*)d_in[2];
  const int*   rows = (const int*)d_in[3];
  const int*   cols = (const int*)d_in[4];
  const int    nnz  = in_sizes[1];

  // grid.x (fast) = 188 row tiles, grid.y (slow) = 64 batch blocks:
  // all row tiles of one batch block share a 10 MB x-slice in the 192 MB L2,
  // so HBM streams x essentially once (~760 MB total traffic).
  dim3 grid((OUT_F + ROW_TILE - 1) / ROW_TILE, BATCHN / BATCH_PER_BLOCK);
  spmm_wmma_kernel<<<grid, TPB, 0, stream>>>(x, w, bias, rows, cols, nnz,
                                             (float*)d_out);
}